// RFNetwork_14611478741339
// MI455X (gfx1250) — compile-verified
//
#include <hip/hip_runtime.h>
#include <hip/hip_bf16.h>

// ---------------------------------------------------------------------------
// RF network scan, MI455X (gfx1250, wave32).
//   out_t = (W_fixed + P_t) @ x_t ;  P_{t+1} = renorm(P_t + l*outer(out_t,x_t))
// in_in_p is dead state (never read by the returned outputs) -> skipped.
//
// Design: ONE persistent kernel, 64 workgroups x 256 threads.
//   * Each WG owns 16 rows of P, resident in LDS for the whole scan (64 KB).
//   * base[t][row] = W_fixed @ x_t for all t precomputed in-kernel with
//     V_WMMA_F32_16X16X4_F32 (fp32 WMMA, reference precision).
//   * Row renorm is WG-local; column renorm uses a deterministic 2-stage
//     cross-WG reduction with a monotonic atomic-counter grid barrier
//     (2 barriers/step, no FP atomics -> bitwise deterministic).
// ---------------------------------------------------------------------------

typedef __attribute__((ext_vector_type(2))) float v2f;
typedef __attribute__((ext_vector_type(8))) float v8f;

#define T_STEPS   128
#define N_DIM     1024
#define NWG       64
#define RPW       16          // rows of P per workgroup
#define BLOCK     256
#define NWAVES    8
#define LMBDA     0.01f
#define MAX_POST  1.0f
#define MAX_PRE   1.0f

// Monotonic grid barrier: counter never resets; barrier k waits for k*NWG.
__device__ __forceinline__ void grid_barrier(unsigned* cnt, unsigned k) {
    __syncthreads();
    if (threadIdx.x == 0) {
        __threadfence();  // release prior global writes (agent scope)
        __hip_atomic_fetch_add(cnt, 1u, __ATOMIC_RELEASE, __HIP_MEMORY_SCOPE_AGENT);
        while (__hip_atomic_load(cnt, __ATOMIC_ACQUIRE, __HIP_MEMORY_SCOPE_AGENT) < k * NWG) {
            __builtin_amdgcn_s_sleep(2);
        }
    }
    __syncthreads();
    __threadfence();      // acquire: invalidate stale near caches before reads
}

__global__ __launch_bounds__(BLOCK, 1)
void rf_scan_persistent(const float* __restrict__ X,     // [T][N] inputs
                        const float* __restrict__ Wf,    // [N][N] out_in_fixed
                        float* __restrict__ out,         // [T][N]
                        unsigned* __restrict__ bar_cnt,  // 1 counter (memset 0)
                        float* __restrict__ colfac,      // [N]
                        float* __restrict__ partial)     // [NWG][N]
{
    __shared__ float P[RPW][N_DIM];             // 64 KB: persistent plastic rows
    __shared__ float base_sh[T_STEPS][RPW];     //  8 KB: W_fixed @ x_t, all t
    __shared__ float x_sh[N_DIM];               //  4 KB
    __shared__ float cfac_sh[N_DIM];            //  4 KB
    __shared__ float o_sh[RPW];
    __shared__ float f_sh[RPW];

    const int tid  = threadIdx.x;
    const int lane = tid & 31;
    const int wave = tid >> 5;
    const int wg   = blockIdx.x;
    const int row0 = wg * RPW;

    // ---- zero P ----
    for (int i = tid; i < RPW * N_DIM; i += BLOCK) (&P[0][0])[i] = 0.0f;

    // ---- base GEMM via fp32 WMMA: D[16 time x 16 row] tiles, K = N_DIM ----
    // A = X tile (m=time, k), B = Wf^T tile (k, n=row). f32 16x16x4 layout:
    // lanes 0-15 hold K={0,1}, lanes 16-31 hold K={2,3} (VGPR0->+0, VGPR1->+1).
    {
        const int t0   = wave * 16;                  // each wave: one time-tile
        const int mn   = lane & 15;                  // A row (time) == B col (row)
        const int koff = (lane >> 4) * 2;
        const float* Arow = X  + (size_t)(t0   + mn) * N_DIM;
        const float* Brow = Wf + (size_t)(row0 + mn) * N_DIM;
        v8f acc = {};
        for (int kk = 0; kk < N_DIM; kk += 4) {
            v2f a = *(const v2f*)(Arow + kk + koff);
            v2f b = *(const v2f*)(Brow + kk + koff);
            acc = __builtin_amdgcn_wmma_f32_16x16x4_f32(
                false, a, false, b, (short)0, acc, false, false);
        }
        const int mbase = (lane >> 4) * 8;           // D: VGPR r -> M = r (+8 hi)
#pragma unroll
        for (int r = 0; r < 8; ++r)
            base_sh[t0 + mbase + r][lane & 15] = acc[r];
    }
    __syncthreads();

    // ---- sequential scan over T steps ----
    unsigned barK = 0;
    for (int t = 0; t < T_STEPS; ++t) {
        // load x_t (hot in L2); prefetch x_{t+1}
        for (int j = tid; j < N_DIM; j += BLOCK) x_sh[j] = X[(size_t)t * N_DIM + j];
        if (t + 1 < T_STEPS)
            __builtin_prefetch(&X[(size_t)(t + 1) * N_DIM + tid * 4], 0, 1);
        __syncthreads();

        // matvec: out[i] = base[t][i] + P[i]·x   (wave w -> rows w, w+8)
        for (int rr = wave; rr < RPW; rr += NWAVES) {
            float s = 0.0f;
            for (int j = lane; j < N_DIM; j += 32) s += P[rr][j] * x_sh[j];
#pragma unroll
            for (int off = 16; off > 0; off >>= 1) s += __shfl_xor(s, off);
            if (lane == 0) {
                float o = base_sh[t][rr] + s;
                o_sh[rr] = o;
                out[(size_t)t * N_DIM + row0 + rr] = o;
            }
        }
        __syncthreads();

        // Hebbian update + row-sum renorm factor (row-local)
        for (int rr = wave; rr < RPW; rr += NWAVES) {
            const float lo = LMBDA * o_sh[rr];
            float rs = 0.0f;
            for (int j = lane; j < N_DIM; j += 32) {
                float v = P[rr][j] + lo * x_sh[j];
                P[rr][j] = v;
                rs += v;
            }
#pragma unroll
            for (int off = 16; off > 0; off >>= 1) rs += __shfl_xor(rs, off);
            if (lane == 0) f_sh[rr] = (rs > MAX_POST) ? (MAX_POST / rs) : 1.0f;
        }
        __syncthreads();

        // per-WG partial column sums of row-scaled P (deterministic order)
        for (int j = tid; j < N_DIM; j += BLOCK) {
            float s = 0.0f;
#pragma unroll
            for (int i = 0; i < RPW; ++i) s += f_sh[i] * P[i][j];
            partial[(size_t)wg * N_DIM + j] = s;
        }

        grid_barrier(bar_cnt, ++barK);               // partials visible

        // this WG finalizes columns [wg*16, wg*16+16)
        if (tid < RPW) {
            const int j = wg * RPW + tid;
            float s = 0.0f;
#pragma unroll 8
            for (int p = 0; p < NWG; ++p) s += partial[(size_t)p * N_DIM + j];
            colfac[j] = (s > MAX_PRE) ? (MAX_PRE / s) : 1.0f;
        }

        grid_barrier(bar_cnt, ++barK);               // colfac visible

        // apply row*col renorm to resident P
        for (int j = tid; j < N_DIM; j += BLOCK) cfac_sh[j] = colfac[j];
        __syncthreads();
        for (int rr = wave; rr < RPW; rr += NWAVES) {
            const float f = f_sh[rr];
            for (int j = lane; j < N_DIM; j += 32) P[rr][j] *= f * cfac_sh[j];
        }
        __syncthreads();
    }
}

extern "C" void kernel_launch(void* const* d_in, const int* in_sizes, int n_in,
                              void* d_out, int out_size, void* d_ws, size_t ws_size,
                              hipStream_t stream) {
    (void)in_sizes; (void)n_in; (void)out_size; (void)ws_size;
    const float* X  = (const float*)d_in[0];   // inputs       [128,1024]
    /* d_in[1] = in_in_fixed: dead state, unused */
    const float* Wf = (const float*)d_in[2];   // out_in_fixed [1024,1024]

    // workspace layout: [0,256) barrier counter | colfac 4KB | partials 256KB
    unsigned* bar_cnt = (unsigned*)d_ws;
    float*    colfac  = (float*)((char*)d_ws + 256);
    float*    partial = (float*)((char*)d_ws + 256 + N_DIM * sizeof(float));

    hipMemsetAsync(d_ws, 0, 256, stream);      // re-init barrier each call
    rf_scan_persistent<<<dim3(NWG), dim3(BLOCK), 0, stream>>>(
        X, Wf, (float*)d_out, bar_cnt, colfac, partial);
}